// TorchNet_36610301231691
// MI455X (gfx1250) — compile-verified
//
#include <hip/hip_runtime.h>

// One wave32 handles 16 nodes. The 32-child sum for those 16 nodes is done as
// D += A(16x4) x ones(4x16) via V_WMMA_F32_16X16X4_F32 (exact fp32), chained 8x.
//
// A-matrix VGPR layout (32-bit A, 16x4, per ISA 7.12.2):
//   lanes 0-15 : M=lane,    VGPR0=K0, VGPR1=K1
//   lanes 16-31: M=lane-16, VGPR0=K2, VGPR1=K3
// So lane (m, half) loads children k = 4*i + 2*half + {0,1} of node (base+m):
// a contiguous int2 of indices -> two gathered f32 values.
//
// D layout: VGPR r holds M=r (lanes 0-15, N=lane) and M=r+8 (lanes 16-31).
// With B == ones every column is identical, so lane 0 / lane 16 broadcast
// rows r / r+8; we redistribute so lane n ends with node base+n's sum.

typedef __attribute__((ext_vector_type(2))) float v2f;
typedef __attribute__((ext_vector_type(8))) float v8f;

__global__ __launch_bounds__(256) void tree_layer_wmma(
    const float* __restrict__ vals_in,   // previous layer values [N]
    float* __restrict__ vals_out,        // this layer values [N]
    const int* __restrict__ ci,          // child indices for this layer [N][32]
    const int* __restrict__ fi,          // activation ids for this layer [N]
    const float* __restrict__ wptr,      // scalar weight
    int n_nodes)
{
    const int lane = (int)threadIdx.x & 31;
    const int wave = ((int)blockIdx.x * (int)blockDim.x + (int)threadIdx.x) >> 5;
    const int nodeBase = wave * 16;
    if (nodeBase >= n_nodes) return;     // wave-uniform: EXEC stays all-ones

    const int m    = lane & 15;          // node within the 16-node tile
    const int half = lane >> 4;          // 0 -> K{0,1}, 1 -> K{2,3}
    int node = nodeBase + m;
    if (node >= n_nodes) node = n_nodes - 1;   // clamp loads, keep EXEC full
    const size_t rowBase = (size_t)node * 32u + (size_t)(2 * half);

    const float w = wptr[0];

    v2f bOnes; bOnes[0] = 1.0f; bOnes[1] = 1.0f;   // B = ones -> D = row sums
    v8f acc = {};                                   // C/D accumulator (8 VGPRs)

    #pragma unroll
    for (int i = 0; i < 8; ++i) {
        // contiguous pair of child indices for this lane's (node, K-slot)
        const int2 p = *(const int2*)(ci + rowBase + (size_t)(4 * i));
        v2f a;
        a[0] = vals_in[p.x];   // gather: 800KB table, L2-resident
        a[1] = vals_in[p.y];
        acc = __builtin_amdgcn_wmma_f32_16x16x4_f32(
            /*neg_a=*/false, a, /*neg_b=*/false, bOnes,
            /*c_mod=*/(short)0, acc, /*reuse_a=*/false, /*reuse_b=*/false);
    }

    // Redistribute: node base+r lives in acc[r] (lane 0) / acc[r]+8 (lane 16).
    float sum = 0.0f;
    #pragma unroll
    for (int r = 0; r < 8; ++r) {
        const float lo = __shfl(acc[r], 0, 32);    // node base+r
        const float hi = __shfl(acc[r], 16, 32);   // node base+r+8
        sum = (lane == r)     ? lo : sum;
        sum = (lane == r + 8) ? hi : sum;
    }

    // Unconditional, clamped fi load: lanes 16-31 duplicate lanes 0-15's
    // entries (same cache lines) -> no divergent-branch/EXEC save needed.
    const int outNode = nodeBase + (lane & 15);
    const int fNode   = (outNode < n_nodes) ? outNode : (n_nodes - 1);
    const int f = fi[fNode];

    const float x = w * sum;
    float y = x;                                   // fid == 3: identity
    if      (f == 0) y = tanhf(x);
    else if (f == 1) y = 1.0f / (1.0f + expf(-x)); // sigmoid
    else if (f == 2) y = fmaxf(x, 0.0f);           // relu

    if (lane < 16 && outNode < n_nodes) vals_out[outNode] = y;
}

extern "C" void kernel_launch(void* const* d_in, const int* in_sizes, int n_in,
                              void* d_out, int out_size, void* d_ws, size_t ws_size,
                              hipStream_t stream) {
    const float* X         = (const float*)d_in[0];
    const float* w         = (const float*)d_in[1];
    const int*   child_idx = (const int*)d_in[2];
    const int*   fun_ids   = (const int*)d_in[3];
    float* out = (float*)d_out;
    float* ws  = (float*)d_ws;

    const int N = in_sizes[0];                     // 200000
    const int L = (N > 0) ? in_sizes[3] / N : 0;   // 8 layers (fun_ids = L*N)
    // kernel assumes K == 32 children (matches reference N_CHILD)

    const int waves   = (N + 15) / 16;
    const int threads = 256;                       // 8 waves / block
    const int blocks  = (waves * 32 + threads - 1) / threads;

    // Back-parity ping-pong: layer L-1 always writes d_out, and no layer
    // reads the buffer it writes (layer 0 reads X).
    const float* cur = X;
    for (int l = 0; l < L; ++l) {
        float* dst = (((L - 1 - l) & 1) != 0) ? ws : out;
        tree_layer_wmma<<<blocks, threads, 0, stream>>>(
            cur, dst,
            child_idx + (size_t)l * (size_t)N * 32u,
            fun_ids   + (size_t)l * (size_t)N,
            w, N);
        cur = dst;
    }
}